// GridManifoldNetwork_44263932952591
// MI455X (gfx1250) — compile-verified
//
#include <hip/hip_runtime.h>

typedef __attribute__((ext_vector_type(2))) float v2f;
typedef __attribute__((ext_vector_type(8))) float v8f;

// Hash-grid constants (tcnn defaults from the reference)
// levels 0..5: res = 16<<l ; sizes: 4096,32768,262144,524288,524288,524288
__device__ __constant__ unsigned c_offs[6] = {0u, 4096u, 36864u, 299008u, 823296u, 1347584u};

__global__ __launch_bounds__(256) void gridnet_fused_kernel(
    const float* __restrict__ X,    // [N,3]
    const float* __restrict__ grid, // [TOTAL,2]
    const float* __restrict__ W0,   // [15,64]
    const float* __restrict__ b0,   // [64]
    const float* __restrict__ W1,   // [64,1]
    const float* __restrict__ b1,   // [1]
    float* __restrict__ out,        // [N]
    int n)
{
    __shared__ float As[256 * 17];   // 16 tiles of [16 pts][K=16] (stride 17: bank-conflict-free)
    __shared__ float W0s[16 * 64];   // K-padded W0, row-major [K][N]
    __shared__ float W1s[64];
    __shared__ float b0s[64];

    const int tid = threadIdx.x;

    // ---- cooperative preamble: stage weights in LDS ----
    for (int i = tid; i < 16 * 64; i += 256)
        W0s[i] = (i < 15 * 64) ? W0[i] : 0.0f;     // zero-pad K=15 row
    if (tid < 64) { W1s[tid] = W1[tid]; b0s[tid] = b0[tid]; }
    const float b1v = b1[0];

    // ---- phase 1: hash-grid encode, one point per thread ----
    const int pt  = blockIdx.x * 256 + tid;
    const int ptc = (pt < n) ? pt : (n - 1);

    const float x0 = X[3 * ptc + 0];
    const float x1 = X[3 * ptc + 1];
    const float x2 = X[3 * ptc + 2];

    float* row = &As[tid * 17];
    row[0] = x0; row[1] = x1; row[2] = x2;
    row[15] = 0.0f;                                 // K pad

    const float xn0 = (x0 + 1.0f) * 0.5f;
    const float xn1 = (x1 + 1.0f) * 0.5f;
    const float xn2 = (x2 + 1.0f) * 0.5f;

#pragma unroll
    for (int l = 0; l < 6; ++l) {
        const unsigned res = 16u << l;
        const float    sc  = (float)(res - 1u);
        const unsigned off = c_offs[l];

        const float px = xn0 * sc + 0.5f;
        const float py = xn1 * sc + 0.5f;
        const float pz = xn2 * sc + 0.5f;
        const float fx = floorf(px), fy = floorf(py), fz = floorf(pz);
        const float wx = px - fx,   wy = py - fy,   wz = pz - fz;
        const unsigned ix = (unsigned)fx, iy = (unsigned)fy, iz = (unsigned)fz;

        float a0 = 0.0f, a1 = 0.0f;
#pragma unroll
        for (int c = 0; c < 8; ++c) {
            const unsigned bx = c & 1u, by = (c >> 1) & 1u, bz = (c >> 2) & 1u;
            unsigned cx = ix + bx, cy = iy + by, cz = iz + bz;
            const float w = (bx ? wx : 1.0f - wx) *
                            (by ? wy : 1.0f - wy) *
                            (bz ? wz : 1.0f - wz);
            unsigned idx;
            if (l < 3) {                             // dense grid
                const unsigned lim = res - 1u;
                cx = cx < lim ? cx : lim;
                cy = cy < lim ? cy : lim;
                cz = cz < lim ? cz : lim;
                idx = cx + cy * res + cz * res * res;
            } else {                                 // hashed, size = 2^19
                idx = (cx ^ (cy * 2654435761u) ^ (cz * 805459861u)) & 0x7FFFFu;
            }
            const float2 g = ((const float2*)grid)[off + idx];   // global_load_b64, L2-resident
            a0 = fmaf(w, g.x, a0);
            a1 = fmaf(w, g.y, a1);
        }
        row[3 + 2 * l] = a0;
        row[4 + 2 * l] = a1;
    }
    __syncthreads();

    // ---- phase 2: fused MLP via V_WMMA_F32_16X16X4_F32 ----
    const int wave = tid >> 5;
    const int lane = tid & 31;
    const int half = lane >> 4;      // 0: M rows v / K 4s..4s+1 ; 1: M rows v+8 / K 4s+2..4s+3
    const int lm   = lane & 15;

    const v8f vzero = {0.f, 0.f, 0.f, 0.f, 0.f, 0.f, 0.f, 0.f};

#pragma unroll
    for (int tt = 0; tt < 2; ++tt) {
        const int tileIdx = wave * 2 + tt;           // 16 point-tiles per block
        const float* Ab = &As[tileIdx * 16 * 17];

        v8f acc[4];
#pragma unroll
        for (int t = 0; t < 4; ++t) acc[t] = vzero;

#pragma unroll
        for (int s = 0; s < 4; ++s) {                // K steps: 4 x K=4
            const int ka = 4 * s + 2 * half;
            // A fragment: 16x4 f32 layout (lane = M, VGPR0/1 = K, K+1; halves split K pairs)
            const float* ap = &Ab[lm * 17 + ka];
            v2f a; a.x = ap[0]; a.y = ap[1];
#pragma unroll
            for (int t = 0; t < 4; ++t) {            // N tiles: hidden dims 16t..16t+15
                v2f b;
                b.x = W0s[(ka + 0) * 64 + t * 16 + lm];
                b.y = W0s[(ka + 1) * 64 + t * 16 + lm];
                acc[t] = __builtin_amdgcn_wmma_f32_16x16x4_f32(
                    /*neg_a=*/false, a, /*neg_b=*/false, b,
                    /*c_mod=*/(short)0, acc[t],
                    /*reuse_a=*/false, /*reuse_b=*/false);
            }
        }

        // ---- layer 1 folded in-register: out[m] = sum_n relu(h+b0)*W1[n] + b1 ----
        float sum[8] = {0.f, 0.f, 0.f, 0.f, 0.f, 0.f, 0.f, 0.f};
#pragma unroll
        for (int t = 0; t < 4; ++t) {
            const float wv = W1s[t * 16 + lm];
            const float bb = b0s[t * 16 + lm];
#pragma unroll
            for (int v = 0; v < 8; ++v) {
                float h = acc[t][v] + bb;            // D layout: M = v + 8*half, N = 16t + lm
                h = h > 0.0f ? h : 0.0f;
                sum[v] = fmaf(h, wv, sum[v]);
            }
        }
        // reduce across the 16 lanes of each half (N dimension)
#pragma unroll
        for (int m = 1; m <= 8; m <<= 1) {
#pragma unroll
            for (int v = 0; v < 8; ++v)
                sum[v] += __shfl_xor(sum[v], m, 32);
        }

        if (lm == 0) {
            const int base = blockIdx.x * 256 + tileIdx * 16 + half * 8;
            if (base + 7 < n) {
                float4 o0 = make_float4(sum[0] + b1v, sum[1] + b1v, sum[2] + b1v, sum[3] + b1v);
                float4 o1 = make_float4(sum[4] + b1v, sum[5] + b1v, sum[6] + b1v, sum[7] + b1v);
                *(float4*)&out[base + 0] = o0;       // global_store_b128 x2
                *(float4*)&out[base + 4] = o1;
            } else {
#pragma unroll
                for (int v = 0; v < 8; ++v)
                    if (base + v < n) out[base + v] = sum[v] + b1v;
            }
        }
    }
}

extern "C" void kernel_launch(void* const* d_in, const int* in_sizes, int n_in,
                              void* d_out, int out_size, void* d_ws, size_t ws_size,
                              hipStream_t stream) {
    const float* X    = (const float*)d_in[0];
    const float* grid = (const float*)d_in[1];
    const float* W0   = (const float*)d_in[2];
    const float* b0   = (const float*)d_in[3];
    const float* W1   = (const float*)d_in[4];
    const float* b1   = (const float*)d_in[5];
    float* out = (float*)d_out;

    const int n = in_sizes[0] / 3;                  // 2,097,152 points
    const int blocks = (n + 255) / 256;
    gridnet_fused_kernel<<<blocks, 256, 0, stream>>>(X, grid, W0, b0, W1, b1, out, n);
}